// HypComEnc_37838661878282
// MI455X (gfx1250) — compile-verified
//
#include <hip/hip_runtime.h>
#include <math.h>

// =====================================================================
// HGCN forward for MI455X (gfx1250, wave32).
//  - Möbius matvec with V_WMMA_F32_16X16X4_F32 (full f32 precision).
//  - Both A and B fragments are staged in LDS (padded stride -> zero
//    bank conflicts); this replaces the previous dependent global_load
//    -> s_wait_loadcnt 0 -> wmma serialization with pipelined ds loads.
//  - Edge aggregation: per-edge wave32 gather + f32 atomic scatter;
//    the 51.2MB node table is L2-resident (192MB L2).
// Workspace: buf0 N*128 f32, buf1 N*128 f32, small bias area (~103 MB).
// =====================================================================

#define MAXN 0.996f   // (1 - 4e-3)/sqrt(c), c = 1
#define MINN 1e-15f
#define KC   128
#define BM   64

typedef float v2f __attribute__((ext_vector_type(2)));
typedef float v8f __attribute__((ext_vector_type(8)));

__device__ __forceinline__ float artanh_f(float z) {
    z = fminf(fmaxf(z, -(1.f - 1e-7f)), 1.f - 1e-7f);
    return 0.5f * logf((1.f + z) / (1.f - z));
}

// ---------------------------------------------------------------------
// bias_ball = proj(expmap0(b)); also stores y2 = ||bias_ball||^2.
// grid.x = 2 (layer 1 / layer 2), 128 threads.
// ---------------------------------------------------------------------
__global__ void bias_prep(const float* __restrict__ b1,
                          const float* __restrict__ b2,
                          float* __restrict__ bb, float* __restrict__ y2) {
    __shared__ float red[128];
    const float* b = blockIdx.x ? b2 : b1;
    int t = threadIdx.x;
    float v = b[t];
    red[t] = v * v;
    __syncthreads();
    for (int s = 64; s > 0; s >>= 1) {
        if (t < s) red[t] += red[t + s];
        __syncthreads();
    }
    float bn = fmaxf(sqrtf(red[0]), MINN);
    float tc = fminf(tanhf(bn), MAXN);
    bb[blockIdx.x * 128 + t] = tc / bn * v;
    if (t == 0) y2[blockIdx.x] = tc * tc;
}

// ---------------------------------------------------------------------
// Fused HypLinear: ht = logmap0(proj(mobius_add(proj(mobius_matvec(W,x)),
//                                               bias_ball)))
// 256-thread block -> 64-row x 128-col tile. 8 waves; wave w:
// row-strip w>>1 (16 rows), col-half w&1 (4 x 16 col-tiles).
// A (pre-scaled by expmap0/proj factor) and B both staged in LDS.
// ---------------------------------------------------------------------
__global__ __launch_bounds__(256) void hyp_linear_wmma(
    const float* __restrict__ X, int Kdim, int nrows,
    const float* __restrict__ W,    // [128 x Kdim] row-major
    const float* __restrict__ bb,   // bias ball point [128]
    const float* __restrict__ y2p,  // ||bias ball||^2
    float* __restrict__ HT, int apply_exp) {
    __shared__ float As[BM][KC + 4];
    __shared__ float Ws[128][KC + 4];
    __shared__ float red[BM][4];
    __shared__ float scale_row[BM], xn_row[BM];
    __shared__ float rs_mx2[BM], rs_dot[BM];
    __shared__ float rowCA[BM], rowCB[BM];

    const int t = threadIdx.x;
    const int row0 = blockIdx.x * BM;

    // ---- phase A: per-row norm stats (+ optional expmap0 scaling) ----
    {
        int r = t >> 2, p = t & 3;
        int gr = row0 + r;
        float s = 0.f;
        if (gr < nrows) {
            int kl = Kdim >> 2;
            const float* px = X + (size_t)gr * Kdim + p * kl;
            for (int k = 0; k < kl; k += 4) {
                float4 v = *(const float4*)(px + k);
                s += v.x * v.x + v.y * v.y + v.z * v.z + v.w * v.w;
            }
        }
        red[r][p] = s;
    }
    if (t < BM) { rs_mx2[t] = 0.f; rs_dot[t] = 0.f; }
    __syncthreads();
    if (t < BM) {
        float un2 = red[t][0] + red[t][1] + red[t][2] + red[t][3];
        float un = fmaxf(sqrtf(un2), MINN);
        if (apply_exp) {
            float bc = fminf(tanhf(un), MAXN);   // ball norm after proj
            scale_row[t] = bc / un;              // expmap0 + proj scale
            xn_row[t] = fmaxf(bc, MINN);
        } else {
            scale_row[t] = 1.f;
            xn_row[t] = un;
        }
    }
    __syncthreads();

    const int lane = t & 31;
    const int wid = t >> 5;      // 0..7
    const int rs = wid >> 1;     // row strip 0..3
    const int ch = wid & 1;      // col half 0..1
    const int half = lane >> 4;  // upper/lower 16 lanes
    const int l15 = lane & 15;

    v8f acc[4];
    for (int j = 0; j < 4; ++j)
        for (int i = 0; i < 8; ++i) acc[j][i] = 0.f;

    const int nchunks = Kdim / KC;

    for (int kc = 0; kc < nchunks; ++kc) {
        // stage 64 x KC A-chunk (pre-scaled to the ball) into LDS
        for (int f = t; f < BM * KC / 4; f += 256) {
            int r = f >> 5;        // KC/4 == 32 float4 per row
            int c4 = f & 31;
            int gr = row0 + r;
            float4 v = make_float4(0.f, 0.f, 0.f, 0.f);
            if (gr < nrows) {
                v = *(const float4*)(X + (size_t)gr * Kdim + kc * KC + c4 * 4);
                float sc = scale_row[r];
                v.x *= sc; v.y *= sc; v.z *= sc; v.w *= sc;
            }
            *(float4*)(&As[r][c4 * 4]) = v;
        }
        // stage 128 x KC W-chunk into LDS (coalesced, shared by all waves)
        for (int f = t; f < 128 * KC / 4; f += 256) {
            int r = f >> 5;
            int c4 = f & 31;
            float4 v = *(const float4*)(W + (size_t)r * Kdim + kc * KC + c4 * 4);
            *(float4*)(&Ws[r][c4 * 4]) = v;
        }
        __syncthreads();

        const int mr = rs * 16 + l15;
        for (int kk = 0; kk < KC; kk += 4) {
            // A frag (16x4 f32): lane=M, vgpr0={K0|K2}, vgpr1={K1|K3}
            const int ko = kk + half * 2;
            v2f a;
            a.x = As[mr][ko];
            a.y = As[mr][ko + 1];
#pragma unroll
            for (int j = 0; j < 4; ++j) {
                // B frag (4x16 f32): lane=N, mirrored K pairing (b64 pair)
                const int colg = (ch * 4 + j) * 16 + l15;
                v2f b;
                b.x = Ws[colg][ko];
                b.y = Ws[colg][ko + 1];
                acc[j] = __builtin_amdgcn_wmma_f32_16x16x4_f32(
                    false, a, false, b, (short)0, acc[j], false, false);
            }
        }
        __syncthreads();
    }

    // ---- epilogue: row scalars ||mx||^2 and <mx, bias_ball> ----
    float bbv[4];
#pragma unroll
    for (int j = 0; j < 4; ++j) bbv[j] = bb[(ch * 4 + j) * 16 + l15];

#pragma unroll
    for (int i = 0; i < 8; ++i) {
        float p2 = 0.f, pd = 0.f;
#pragma unroll
        for (int j = 0; j < 4; ++j) {
            float v = acc[j][i];
            p2 += v * v;
            pd += v * bbv[j];
        }
        // C/D layout: vgpr i, lanes0-15 -> M=i, lanes16-31 -> M=i+8
        for (int m = 1; m < 16; m <<= 1) {
            p2 += __shfl_xor(p2, m, 16);
            pd += __shfl_xor(pd, m, 16);
        }
        int mrow = rs * 16 + i + half * 8;
        if (l15 == 0) {
            atomicAdd(&rs_mx2[mrow], p2);
            atomicAdd(&rs_dot[mrow], pd);
        }
    }
    __syncthreads();

    // analytic mobius_matvec + proj + mobius_add + proj + logmap0 scalars
    if (t < BM) {
        float xn = xn_row[t];
        float mxn = fmaxf(sqrtf(rs_mx2[t]), MINN);
        float tv = tanhf(mxn / xn * artanh_f(xn));
        float tc = fminf(tv, MAXN);       // proj of mv norm
        float s_mv = tc / mxn;            // mv_ball = s_mv * mx_raw
        float x2 = tc * tc;
        float xy = s_mv * rs_dot[t];
        float y2 = y2p[0];
        float al = 1.f + 2.f * xy + y2;
        float be = 1.f - x2;
        float den = fmaxf(1.f + 2.f * xy + x2 * y2, MINN);
        float nn2 = fmaxf(al * al * x2 + 2.f * al * be * xy + be * be * y2, 0.f);
        float hn = fmaxf(sqrtf(nn2) / den, MINN);
        float hc = fminf(hn, MAXN);       // proj
        float L = artanh_f(hc) / hn;      // logmap0 (+proj scale) per row
        rowCA[t] = L * al * s_mv / den;
        rowCB[t] = L * be / den;
    }
    __syncthreads();

#pragma unroll
    for (int i = 0; i < 8; ++i) {
        int mrow = rs * 16 + i + half * 8;
        int gr = row0 + mrow;
        if (gr < nrows) {
            float ca = rowCA[mrow], cb = rowCB[mrow];
#pragma unroll
            for (int j = 0; j < 4; ++j) {
                int colg = (ch * 4 + j) * 16 + l15;
                HT[(size_t)gr * 128 + colg] = ca * acc[j][i] + cb * bbv[j];
            }
        }
    }
}

// ---------------------------------------------------------------------
// agg[row[e]] += ht[col[e]]  -- one wave32 per edge, 4 floats per lane.
// ---------------------------------------------------------------------
__global__ __launch_bounds__(256) void edge_agg(const float* __restrict__ HT,
                                                const int* __restrict__ erow,
                                                const int* __restrict__ ecol,
                                                float* __restrict__ AGG, int E) {
    int e = blockIdx.x * 8 + (threadIdx.x >> 5);
    if (e >= E) return;
    int lane = threadIdx.x & 31;
    int r = erow[e], c = ecol[e];
    float4 v = *(const float4*)(HT + (size_t)c * 128 + lane * 4);
    float* dst = AGG + (size_t)r * 128 + lane * 4;
    atomicAdd(dst + 0, v.x);
    atomicAdd(dst + 1, v.y);
    atomicAdd(dst + 2, v.z);
    atomicAdd(dst + 3, v.w);
}

// ---------------------------------------------------------------------
// in-place: h = proj(expmap0(tanh(logmap0(proj(expmap0(agg))))))
// ---------------------------------------------------------------------
__global__ __launch_bounds__(256) void post_agg(float* __restrict__ H, int nrows) {
    int r = blockIdx.x * 8 + (threadIdx.x >> 5);
    if (r >= nrows) return;
    int lane = threadIdx.x & 31;
    float* p = H + (size_t)r * 128 + lane * 4;
    float4 v = *(const float4*)p;
    float an2 = v.x * v.x + v.y * v.y + v.z * v.z + v.w * v.w;
    for (int m = 1; m < 32; m <<= 1) an2 += __shfl_xor(an2, m, 32);
    float an = fmaxf(sqrtf(an2), MINN);
    float tn = fminf(fmaxf(tanhf(an), MINN), MAXN);  // proj'd ball norm
    float lg = artanh_f(tn) / an;                    // logmap0 scale
    float4 at;
    at.x = tanhf(lg * v.x);
    at.y = tanhf(lg * v.y);
    at.z = tanhf(lg * v.z);
    at.w = tanhf(lg * v.w);
    float vn2 = at.x * at.x + at.y * at.y + at.z * at.z + at.w * at.w;
    for (int m = 1; m < 32; m <<= 1) vn2 += __shfl_xor(vn2, m, 32);
    float vn = fmaxf(sqrtf(vn2), MINN);
    float s = fminf(tanhf(vn), MAXN) / vn;           // expmap0 + proj
    *(float4*)p = make_float4(s * at.x, s * at.y, s * at.z, s * at.w);
}

// ---------------------------------------------------------------------
// ht = logmap0(proj(h)); ssum[seg] += ht; cnt[seg] += 1
// ---------------------------------------------------------------------
__global__ __launch_bounds__(256) void final_scatter(
    const float* __restrict__ H, const int* __restrict__ seg,
    float* __restrict__ ssum, float* __restrict__ cnt, int nrows) {
    int r = blockIdx.x * 8 + (threadIdx.x >> 5);
    if (r >= nrows) return;
    int lane = threadIdx.x & 31;
    float4 v = *(const float4*)(H + (size_t)r * 128 + lane * 4);
    float hn2 = v.x * v.x + v.y * v.y + v.z * v.z + v.w * v.w;
    for (int m = 1; m < 32; m <<= 1) hn2 += __shfl_xor(hn2, m, 32);
    float hn = fmaxf(sqrtf(hn2), MINN);
    float pc = fminf(hn, MAXN);
    float L = artanh_f(pc) / hn;
    int s = seg[r];
    float* d = ssum + (size_t)s * 128 + lane * 4;
    atomicAdd(d + 0, L * v.x);
    atomicAdd(d + 1, L * v.y);
    atomicAdd(d + 2, L * v.z);
    atomicAdd(d + 3, L * v.w);
    if (lane == 0) atomicAdd(&cnt[s], 1.f);
}

// ---------------------------------------------------------------------
// out = proj(expmap0(ssum / max(cnt,1)))  -- one wave per segment
// ---------------------------------------------------------------------
__global__ __launch_bounds__(256) void pool_k(const float* __restrict__ ssum,
                                              const float* __restrict__ cnt,
                                              float* __restrict__ out, int nseg) {
    int r = blockIdx.x * 8 + (threadIdx.x >> 5);
    if (r >= nseg) return;
    int lane = threadIdx.x & 31;
    float inv = 1.f / fmaxf(cnt[r], 1.f);
    float4 v = *(const float4*)(ssum + (size_t)r * 128 + lane * 4);
    v.x *= inv; v.y *= inv; v.z *= inv; v.w *= inv;
    float mn2 = v.x * v.x + v.y * v.y + v.z * v.z + v.w * v.w;
    for (int m = 1; m < 32; m <<= 1) mn2 += __shfl_xor(mn2, m, 32);
    float mn = fmaxf(sqrtf(mn2), MINN);
    float s = fminf(tanhf(mn), MAXN) / mn;
    float* o = out + (size_t)r * 128 + lane * 4;
    o[0] = s * v.x; o[1] = s * v.y; o[2] = s * v.z; o[3] = s * v.w;
}

__global__ void zero_f32(float* __restrict__ p, size_t n) {
    size_t i = (size_t)blockIdx.x * blockDim.x + threadIdx.x;
    size_t st = (size_t)gridDim.x * blockDim.x;
    for (; i < n; i += st) p[i] = 0.f;
}

extern "C" void kernel_launch(void* const* d_in, const int* in_sizes, int n_in,
                              void* d_out, int out_size, void* d_ws, size_t ws_size,
                              hipStream_t stream) {
    const float* x = (const float*)d_in[0];
    const int* erow = (const int*)d_in[1];
    const int* ecol = (const int*)d_in[2];
    const int* seg = (const int*)d_in[3];
    const float* W1 = (const float*)d_in[4];
    const float* b1 = (const float*)d_in[5];
    const float* W2 = (const float*)d_in[6];
    const float* b2 = (const float*)d_in[7];

    const int N = in_sizes[0] / 256;
    const int E = in_sizes[1];
    const int NSEG = out_size / 128;

    char* ws = (char*)d_ws;
    size_t bufBytes = (size_t)N * 128 * sizeof(float);
    float* buf0 = (float*)ws;                         // ht (layer out)
    float* buf1 = (float*)(ws + bufBytes);            // agg / ball
    float* bbias = (float*)(ws + 2 * bufBytes);       // 2 x 128
    float* y2 = bbias + 256;                          // 2
    float* ssum = buf0;                               // reused after layer 2 GEMM
    float* cnt = buf0 + (size_t)NSEG * 128;

    dim3 blk(256);
    int gRows = (N + 7) / 8;
    int gGemm = (N + BM - 1) / BM;
    int gEdge = (E + 7) / 8;

    bias_prep<<<2, 128, 0, stream>>>(b1, b2, bbias, y2);

    // ---- layer 1 (K = 256, expmap0 fused into A staging) ----
    hyp_linear_wmma<<<gGemm, blk, 0, stream>>>(x, 256, N, W1, bbias, y2, buf0, 1);
    zero_f32<<<2048, 256, 0, stream>>>(buf1, (size_t)N * 128);
    edge_agg<<<gEdge, blk, 0, stream>>>(buf0, erow, ecol, buf1, E);
    post_agg<<<gRows, blk, 0, stream>>>(buf1, N);

    // ---- layer 2 (K = 128, input already on the ball) ----
    hyp_linear_wmma<<<gGemm, blk, 0, stream>>>(buf1, 128, N, W2, bbias + 128,
                                               y2 + 1, buf0, 0);
    zero_f32<<<2048, 256, 0, stream>>>(buf1, (size_t)N * 128);
    edge_agg<<<gEdge, blk, 0, stream>>>(buf0, erow, ecol, buf1, E);
    post_agg<<<gRows, blk, 0, stream>>>(buf1, N);

    // ---- readout: logmap0 + segment mean + expmap0/proj ----
    zero_f32<<<64, 256, 0, stream>>>(ssum, (size_t)NSEG * 128 + NSEG);
    final_scatter<<<gRows, blk, 0, stream>>>(buf1, seg, ssum, cnt, N);
    pool_k<<<(NSEG + 7) / 8, blk, 0, stream>>>(ssum, cnt, (float*)d_out, NSEG);
}